// InterfaceBoundaryLoss_40492951666750
// MI455X (gfx1250) — compile-verified
//
#include <hip/hip_runtime.h>
#include <hip/hip_bf16.h>
#include <math.h>

// ---------------- problem constants (from the reference) ----------------
constexpr int   Bc      = 64;
constexpr int   Hc      = 1024;
constexpr int   Wc      = 1024;
constexpr float EPS1c   = 80.0f;
constexpr float EPS2c   = 2.0f;
constexpr float DXc     = 0.001f;
constexpr float DYc     = 0.001f;
constexpr float CXc     = 512.0f;
constexpr float CYc     = 512.0f;
constexpr float WEIGHTc = 1.0f;

constexpr int NPIX    = Hc * Wc;        // 1,048,576
constexpr int TPB     = 256;            // 8 waves (wave32)
constexpr int NBLK_A  = NPIX / TPB;     // 4096 mask blocks
constexpr int NBLK_D  = 1024;           // reduction blocks

typedef __attribute__((ext_vector_type(2))) float v2f;
typedef __attribute__((ext_vector_type(8))) float v8f;

// ---------------------------------------------------------------------------
// Wave32 cross-lane reduction of two independent f32 accumulators (pot, der)
// using V_WMMA_F32_16X16X4_F32.
//
// A (16x4 f32, 2 VGPRs/lane): lane L<16  -> A[L][0]=a.x,    A[L][1]=a.y
//                             lane L>=16 -> A[L-16][2]=a.x, A[L-16][3]=a.y
// We put pot in a.x, der in a.y for every lane.
// B (4x16 f32): b.x holds rows K=0 (lanes 0-15) / K=2 (lanes 16-31),
//               b.y holds rows K=1 / K=3.
// Selector: B[0][0]=B[2][0]=1 (column 0 sums K=0,2 -> pot),
//           B[1][1]=B[3][1]=1 (column 1 sums K=1,3 -> der).
// D[m][0] = pot(lane m) + pot(lane m+16); D[m][1] = der(m) + der(m+16).
// D layout: lane n<16 VGPR r = D[r][n]; lane n>=16 VGPR r = D[8+r][n-16].
// So column 0 lives in lanes {0,16}, column 1 in lanes {1,17}; each holds 8
// row values -> fold with 8 adds, then 4 readlanes.
// EXEC must be all ones: call only from uniform control flow, full blocks.
// ---------------------------------------------------------------------------
__device__ __forceinline__ void wave_reduce2(float pot, float der,
                                             float& pot_w, float& der_w) {
  const int lane = (int)(threadIdx.x & 31u);
#if defined(__gfx1250__) && __has_builtin(__builtin_amdgcn_wmma_f32_16x16x4_f32)
  v2f a; a.x = pot; a.y = der;
  v2f b;
  b.x = (lane == 0 || lane == 16) ? 1.0f : 0.0f;
  b.y = (lane == 1 || lane == 17) ? 1.0f : 0.0f;
  v8f c = {0.f, 0.f, 0.f, 0.f, 0.f, 0.f, 0.f, 0.f};
  c = __builtin_amdgcn_wmma_f32_16x16x4_f32(
      /*neg_a=*/false, a, /*neg_b=*/false, b,
      /*c_mod=*/(short)0, c, /*reuse_a=*/false, /*reuse_b=*/false);
  float s = c[0] + c[1] + c[2] + c[3] + c[4] + c[5] + c[6] + c[7];
  pot_w = __shfl(s, 0, 32) + __shfl(s, 16, 32);
  der_w = __shfl(s, 1, 32) + __shfl(s, 17, 32);
#else
  // portable wave32 shuffle fallback (also used by the host pass)
  for (int off = 16; off > 0; off >>= 1) {
    pot += __shfl_down(pot, off, 32);
    der += __shfl_down(der, off, 32);
  }
  pot_w = __shfl(pot, 0, 32);
  der_w = __shfl(der, 0, 32);
#endif
}

// ---- kernel 1: per-block masked-pixel counts (ballot + popcount) ----------
__global__ void k_count(const unsigned char* __restrict__ mask,
                        int* __restrict__ counts) {
  const int pix = blockIdx.x * TPB + threadIdx.x;
  const int m = (pix < NPIX) ? (mask[pix] != 0) : 0;
  const unsigned long long bal = __ballot(m);
  __shared__ int wc[TPB / 32];
  const int lane = threadIdx.x & 31, wid = threadIdx.x >> 5;
  if (lane == 0) wc[wid] = __popcll(bal);
  __syncthreads();
  if (threadIdx.x == 0) {
    int t = 0;
    for (int i = 0; i < TPB / 32; ++i) t += wc[i];
    counts[blockIdx.x] = t;
  }
}

// ---- kernel 2: single-block exclusive scan over NBLK_A counts -------------
__global__ void k_scan(const int* __restrict__ counts,
                       int* __restrict__ offsets,
                       int* __restrict__ n_mask) {
  __shared__ int tot[TPB];
  const int t = threadIdx.x;
  const int per = NBLK_A / TPB;        // 16
  int local[16];
  int s = 0;
  const int base = t * per;
  for (int i = 0; i < per; ++i) { local[i] = s; s += counts[base + i]; }
  tot[t] = s;
  __syncthreads();
  for (int off = 1; off < TPB; off <<= 1) {   // Hillis–Steele inclusive scan
    int v = (t >= off) ? tot[t - off] : 0;
    __syncthreads();
    tot[t] += v;
    __syncthreads();
  }
  const int excl = (t == 0) ? 0 : tot[t - 1];
  for (int i = 0; i < per; ++i) offsets[base + i] = excl + local[i];
  if (t == TPB - 1) *n_mask = tot[TPB - 1];
}

// ---- kernel 3: compact masked pixel indices (deterministic, index order) --
__global__ void k_fill(const unsigned char* __restrict__ mask,
                       const int* __restrict__ offsets,
                       unsigned int* __restrict__ list, int cap) {
  const int pix = blockIdx.x * TPB + threadIdx.x;
  const int m = (pix < NPIX) ? (mask[pix] != 0) : 0;
  const unsigned long long bal = __ballot(m);
  __shared__ int wc[TPB / 32];
  const int lane = threadIdx.x & 31, wid = threadIdx.x >> 5;
  if (lane == 0) wc[wid] = __popcll(bal);
  __syncthreads();
  int waveoff = 0;
  for (int i = 0; i < wid; ++i) waveoff += wc[i];
  if (m) {
    const int pos = offsets[blockIdx.x] + waveoff +
                    __popcll(bal & ((1ull << lane) - 1ull));
    if (pos < cap) list[pos] = (unsigned int)pix;
  }
}

// ---- kernel 4: gather 5-point stencils for (pixel, batch) items, reduce ---
__global__ void k_reduce(const float* __restrict__ phi1,
                         const float* __restrict__ phi2,
                         const unsigned int* __restrict__ list,
                         const int* __restrict__ n_mask_p,
                         float* __restrict__ partials, int cap) {
  int nm = *n_mask_p;
  if (nm > cap) nm = cap;
  const unsigned int total  = (unsigned int)nm * (unsigned int)Bc;
  const unsigned int stride = gridDim.x * (unsigned int)TPB;

  float pot = 0.0f, der = 0.0f;
  for (unsigned int idx = blockIdx.x * (unsigned int)TPB + threadIdx.x;
       idx < total; idx += stride) {
    const unsigned int p = idx % (unsigned int)nm;
    const unsigned int b = idx / (unsigned int)nm;
    const unsigned int pix = list[p];
    const int i = (int)(pix >> 10);        // W = 1024
    const int j = (int)(pix & 1023u);
    const size_t base = (size_t)b << 20;   // H*W = 2^20
    const float* __restrict__ a1 = phi1 + base;
    const float* __restrict__ a2 = phi2 + base;

    // edge-replicated central differences (pad mode='edge' after diff)
    int jc = j; if (jc < 1) jc = 1; if (jc > Wc - 2) jc = Wc - 2;
    int ic = i; if (ic < 1) ic = 1; if (ic > Hc - 2) ic = Hc - 2;

    const size_t rc = (size_t)i * Wc;
    const float c1 = a1[rc + j];
    const float c2 = a2[rc + j];
    const float dpx1 = (a1[rc + jc + 1] - a1[rc + jc - 1]) / (2.0f * DXc);
    const float dpx2 = (a2[rc + jc + 1] - a2[rc + jc - 1]) / (2.0f * DXc);
    const float dpy1 = (a1[(size_t)(ic + 1) * Wc + j] -
                        a1[(size_t)(ic - 1) * Wc + j]) / (2.0f * DYc);
    const float dpy2 = (a2[(size_t)(ic + 1) * Wc + j] -
                        a2[(size_t)(ic - 1) * Wc + j]) / (2.0f * DYc);

    const float nx0 = (float)j - CXc;
    const float ny0 = (float)i - CYc;
    const float nrm = sqrtf(nx0 * nx0 + ny0 * ny0);
    const float inv = (nrm > 0.0f) ? (1.0f / nrm) : 1.0f;
    const float nx = nx0 * inv, ny = ny0 * inv;

    const float d1 = nx * dpx1 + ny * dpy1;
    const float d2 = nx * dpx2 + ny * dpy2;
    const float mm = EPS1c * d1 - EPS2c * d2;
    const float dd = c1 - c2;
    pot += dd * dd;
    der += mm * mm;
  }

  // uniform point: all 256 threads reach here -> EXEC all ones for WMMA
  float pw, dw;
  wave_reduce2(pot, der, pw, dw);

  __shared__ float sp[TPB / 32], sd[TPB / 32];
  const int lane = threadIdx.x & 31, wid = threadIdx.x >> 5;
  if (lane == 0) { sp[wid] = pw; sd[wid] = dw; }
  __syncthreads();
  if (threadIdx.x == 0) {
    float P = 0.0f, D = 0.0f;
    for (int i = 0; i < TPB / 32; ++i) { P += sp[i]; D += sd[i]; }
    partials[2 * blockIdx.x]     = P;
    partials[2 * blockIdx.x + 1] = D;
  }
}

// ---- kernel 5: fold block partials, finalize scalar -----------------------
__global__ void k_final(const float* __restrict__ partials,
                        const int* __restrict__ n_mask_p,
                        float* __restrict__ out, int nblk) {
  float pot = 0.0f, der = 0.0f;
  for (int i = threadIdx.x; i < nblk; i += TPB) {
    pot += partials[2 * i];
    der += partials[2 * i + 1];
  }
  float pw, dw;
  wave_reduce2(pot, der, pw, dw);
  __shared__ float sp[TPB / 32], sd[TPB / 32];
  const int lane = threadIdx.x & 31, wid = threadIdx.x >> 5;
  if (lane == 0) { sp[wid] = pw; sd[wid] = dw; }
  __syncthreads();
  if (threadIdx.x == 0) {
    float P = 0.0f, D = 0.0f;
    for (int i = 0; i < TPB / 32; ++i) { P += sp[i]; D += sd[i]; }
    const float denom = (float)Bc * (float)(*n_mask_p);
    out[0] = WEIGHTc * (P + D) / denom;
  }
}

// ---------------------------------------------------------------------------
extern "C" void kernel_launch(void* const* d_in, const int* in_sizes, int n_in,
                              void* d_out, int out_size, void* d_ws,
                              size_t ws_size, hipStream_t stream) {
  const float*         phi1 = (const float*)d_in[0];         // output_in  (B,1,H,W) f32
  const float*         phi2 = (const float*)d_in[1];         // output_out (B,1,H,W) f32
  const unsigned char* mask = (const unsigned char*)d_in[2]; // bool (H,W), 1 byte each

  // workspace layout (4-byte units):
  // [0]            n_mask
  // [1 .. 4096]    per-block counts
  // [4097 .. 8192] per-block offsets
  // [8193 .. ]     2*NBLK_D float partials
  // [...]          compacted pixel index list (remaining capacity)
  int* ws_i          = (int*)d_ws;
  int* n_mask        = ws_i;
  int* counts        = ws_i + 1;
  int* offsets       = ws_i + 1 + NBLK_A;
  float* partials    = (float*)(ws_i + 1 + 2 * NBLK_A);
  unsigned int* list = (unsigned int*)(ws_i + 1 + 2 * NBLK_A + 2 * NBLK_D);

  const size_t used = (size_t)(1 + 2 * NBLK_A + 2 * NBLK_D) * 4;
  long long capll = (ws_size > used) ? (long long)((ws_size - used) / 4) : 0;
  int cap = (capll > (long long)NPIX) ? NPIX : (int)capll;

  k_count <<<NBLK_A, TPB, 0, stream>>>(mask, counts);
  k_scan  <<<1,      TPB, 0, stream>>>(counts, offsets, n_mask);
  k_fill  <<<NBLK_A, TPB, 0, stream>>>(mask, offsets, list, cap);
  k_reduce<<<NBLK_D, TPB, 0, stream>>>(phi1, phi2, list, n_mask, partials, cap);
  k_final <<<1,      TPB, 0, stream>>>(partials, n_mask, (float*)d_out, NBLK_D);
}